// MockTorchStudent_64476049047782
// MI455X (gfx1250) — compile-verified
//
#include <hip/hip_runtime.h>

// GCNConv (self-loops, symmetric norm, edge weights) + global_add_pool + Linear
// Sizes fixed by the reference.
#define N_NODES   100000
#define N_EDGES   600000
#define IN_UNITS  128
#define HID       128
#define OUT_UNITS 64
#define NUM_GRAPHS 128

typedef __attribute__((ext_vector_type(2))) float v2f;
typedef __attribute__((ext_vector_type(8))) float v8f;

// ---------------------------------------------------------------------------
// 0) deg[i] = 1.0 (self-loop weight), pooled = 0
__global__ void gcn_fill_kernel(float* __restrict__ deg, float* __restrict__ pooled) {
    int i = blockIdx.x * blockDim.x + threadIdx.x;
    if (i < N_NODES) deg[i] = 1.0f;
    if (i < NUM_GRAPHS * HID) pooled[i] = 0.0f;
}

// 1) deg[col[e]] += ew[e]
__global__ void gcn_deg_kernel(const int* __restrict__ col,
                               const float* __restrict__ ew,
                               float* __restrict__ deg) {
    int e = blockIdx.x * blockDim.x + threadIdx.x;
    if (e < N_EDGES) atomicAdd(&deg[col[e]], ew[e]);
}

// 2) dis[i] = deg > 0 ? rsqrt(deg) : 0
__global__ void gcn_dis_kernel(const float* __restrict__ deg, float* __restrict__ dis) {
    int i = blockIdx.x * blockDim.x + threadIdx.x;
    if (i < N_NODES) {
        float d = deg[i];
        dis[i] = d > 0.0f ? rsqrtf(d) : 0.0f;
    }
}

// ---------------------------------------------------------------------------
// 3) X = A @ W   via V_WMMA_F32_16X16X4_F32 (fp32 matrix core path, wave32)
//    Block = 256 threads = 8 waves; wave w computes 16x16 tile at cols w*16,
//    block b computes rows b*16. 100000 % 16 == 0 so EXEC stays all-ones.
//
//    A frag (16x4, 2 VGPRs/lane): lane L -> M = L%16, K = (L/16)*2 + v
//    B frag (4x16, 2 VGPRs/lane): lane L -> N = L%16, K = (L/16)*2 + v
//    C/D    (16x16, 8 VGPRs):     lane L -> N = L%16, M = r + (L/16)*8
__global__ __launch_bounds__(256)
void gcn_gemm_wmma_kernel(const float* __restrict__ A,   // [N_NODES, IN_UNITS]
                          const float* __restrict__ W,   // [IN_UNITS, HID]
                          float* __restrict__ X) {       // [N_NODES, HID]
    const int wave = threadIdx.x >> 5;
    const int lane = threadIdx.x & 31;
    const int m0 = blockIdx.x * 16;
    const int n0 = wave * 16;
    const int lm = lane & 15;   // 0..15
    const int lh = lane >> 4;   // 0 or 1

    v8f acc = {};
    const float* arow = A + (size_t)(m0 + lm) * IN_UNITS + lh * 2; // K pair base
    const float* bcol = W + (size_t)(lh * 2) * HID + n0 + lm;      // K pair base

#pragma unroll 4
    for (int k0 = 0; k0 < IN_UNITS; k0 += 4) {
        // consecutive-K pair -> single 64-bit load
        v2f a = *(const v2f*)(arow + k0);
        v2f b;
        b.x = bcol[(size_t)k0 * HID];
        b.y = bcol[(size_t)(k0 + 1) * HID];
        acc = __builtin_amdgcn_wmma_f32_16x16x4_f32(
            /*neg_a=*/false, a, /*neg_b=*/false, b,
            /*c_mod=*/(short)0, acc, /*reuse_a=*/false, /*reuse_b=*/false);
    }

    float* xout = X + n0 + lm;
#pragma unroll
    for (int r = 0; r < 8; ++r) {
        xout[(size_t)(m0 + r + lh * 8) * HID] = acc[r];
    }
}

// ---------------------------------------------------------------------------
// 4) h[i] = dis[i]^2 * x[i] + b_conv   (self-loop term, overwrites h)
__global__ void gcn_selfloop_kernel(const float* __restrict__ x,
                                    const float* __restrict__ dis,
                                    const float* __restrict__ b_conv,
                                    float* __restrict__ h) {
    int t = blockIdx.x * blockDim.x + threadIdx.x;   // over N_NODES*HID
    if (t < N_NODES * HID) {
        int i = t >> 7;          // /128
        int j = t & (HID - 1);
        float s = dis[i];
        h[t] = s * s * x[t] + b_conv[j];
    }
}

// 5) edge scatter: h[col] += dis[row]*ew*dis[col] * x[row]
//    One thread per (edge, 4-float quad): float4 gather + 4 f32 atomics.
__global__ void gcn_scatter_kernel(const int* __restrict__ row,
                                   const int* __restrict__ col,
                                   const float* __restrict__ ew,
                                   const float* __restrict__ dis,
                                   const float* __restrict__ x,
                                   float* __restrict__ h) {
    int t = blockIdx.x * blockDim.x + threadIdx.x;   // E*32 = 19.2M < 2^31
    int e = t >> 5;
    int q = t & 31;
    if (e < N_EDGES) {
        int r = row[e], c = col[e];
        float coeff = dis[r] * ew[e] * dis[c];
        const float4 xv = *(const float4*)(x + (size_t)r * HID + q * 4);
        float* hp = h + (size_t)c * HID + q * 4;
        __builtin_prefetch(hp, 1, 2);   // warm destination line in GL2 (global_prefetch_b8)
        atomicAdd(hp + 0, coeff * xv.x);
        atomicAdd(hp + 1, coeff * xv.y);
        atomicAdd(hp + 2, coeff * xv.z);
        atomicAdd(hp + 3, coeff * xv.w);
    }
}

// 6) pooled[batch[i]] += h[i]
__global__ void gcn_pool_kernel(const float* __restrict__ h,
                                const int* __restrict__ batch,
                                float* __restrict__ pooled) {
    int t = blockIdx.x * blockDim.x + threadIdx.x;   // N_NODES*32 = 3.2M
    int i = t >> 5;
    int q = t & 31;
    if (i < N_NODES) {
        int g = batch[i];
        const float4 hv = *(const float4*)(h + (size_t)i * HID + q * 4);
        float* pp = pooled + (size_t)g * HID + q * 4;
        atomicAdd(pp + 0, hv.x);
        atomicAdd(pp + 1, hv.y);
        atomicAdd(pp + 2, hv.z);
        atomicAdd(pp + 3, hv.w);
    }
}

// 7) out = pooled @ W_lin + b_lin   (128x128x64 = 2 MFLOP, negligible)
__global__ void gcn_linear_kernel(const float* __restrict__ pooled,
                                  const float* __restrict__ W,
                                  const float* __restrict__ b,
                                  float* __restrict__ out) {
    int t = blockIdx.x * blockDim.x + threadIdx.x;
    if (t < NUM_GRAPHS * OUT_UNITS) {
        int g = t / OUT_UNITS;
        int o = t - g * OUT_UNITS;
        float s = b[o];
#pragma unroll 8
        for (int j = 0; j < HID; ++j)
            s += pooled[g * HID + j] * W[j * OUT_UNITS + o];
        out[t] = s;
    }
}

// ---------------------------------------------------------------------------
extern "C" void kernel_launch(void* const* d_in, const int* in_sizes, int n_in,
                              void* d_out, int out_size, void* d_ws, size_t ws_size,
                              hipStream_t stream) {
    const float* node_input = (const float*)d_in[0];   // [100000,128]
    const float* edge_input = (const float*)d_in[1];   // [600000]
    const int*   edge_index = (const int*)d_in[2];     // [2,600000] (int32 per harness)
    const int*   batch      = (const int*)d_in[3];     // [100000]
    const float* W_conv     = (const float*)d_in[4];   // [128,128]
    const float* b_conv     = (const float*)d_in[5];   // [128]
    const float* W_lin      = (const float*)d_in[6];   // [128,64]
    const float* b_lin      = (const float*)d_in[7];   // [64]
    float* out = (float*)d_out;                        // [128,64]

    const int* row = edge_index;            // edge_index[0,:]
    const int* col = edge_index + N_EDGES;  // edge_index[1,:]

    // Workspace layout (all 16B aligned): ~103 MB total
    float* deg    = (float*)d_ws;                       // N_NODES
    float* dis    = deg + N_NODES;                      // N_NODES
    float* x      = dis + N_NODES;                      // N_NODES*HID
    float* h      = x   + (size_t)N_NODES * HID;        // N_NODES*HID
    float* pooled = h   + (size_t)N_NODES * HID;        // NUM_GRAPHS*HID

    const int T = 256;
    gcn_fill_kernel    <<<(N_NODES + T - 1) / T, T, 0, stream>>>(deg, pooled);
    gcn_deg_kernel     <<<(N_EDGES + T - 1) / T, T, 0, stream>>>(col, edge_input, deg);
    gcn_dis_kernel     <<<(N_NODES + T - 1) / T, T, 0, stream>>>(deg, dis);
    gcn_gemm_wmma_kernel<<<N_NODES / 16, T, 0, stream>>>(node_input, W_conv, x);
    gcn_selfloop_kernel<<<(N_NODES * HID + T - 1) / T, T, 0, stream>>>(x, dis, b_conv, h);
    gcn_scatter_kernel <<<(N_EDGES * 32 + T - 1) / T, T, 0, stream>>>(row, col, edge_input, dis, x, h);
    gcn_pool_kernel    <<<(N_NODES * 32 + T - 1) / T, T, 0, stream>>>(h, batch, pooled);
    gcn_linear_kernel  <<<(NUM_GRAPHS * OUT_UNITS + T - 1) / T, T, 0, stream>>>(pooled, W_lin, b_lin, out);
}